// GraphSAGE_3951369912898
// MI455X (gfx1250) — compile-verified
//
#include <hip/hip_runtime.h>

typedef __attribute__((ext_vector_type(2))) float v2f;
typedef __attribute__((ext_vector_type(8))) float v8f;
typedef __attribute__((address_space(3))) float* lds_fp;

#define Nn   50000
#define Ee   800000
#define Dd   128
#define Gg   512
#define OUTd 64
#define EPSf 1e-5f

// ---------------- degree ----------------
__global__ void deg_kernel(const int* __restrict__ dst, float* __restrict__ deg) {
    int e = blockIdx.x * 256 + threadIdx.x;
    if (e < Ee) atomicAdd(&deg[dst[e]], 1.0f);
}

// ---------------- edge scatter: agg[dst] += h[src] ----------------
__global__ void scatter_kernel(const float* __restrict__ hin, const int* __restrict__ src,
                               const int* __restrict__ dst, float* __restrict__ agg) {
    long gid = (long)blockIdx.x * 256 + threadIdx.x;
    int e    = (int)(gid >> 5);
    int lane = (int)(gid & 31);
    if (e >= Ee) return;
    int s = src[e], d = dst[e];
    const float4 v = *reinterpret_cast<const float4*>(hin + (long)s * Dd + lane * 4);
    float* p = agg + (long)d * Dd + lane * 4;
    atomicAdd(p + 0, v.x); atomicAdd(p + 1, v.y);
    atomicAdd(p + 2, v.z); atomicAdd(p + 3, v.w);
}

// ---------------- weight packing into WMMA B-fragment order ----------------
// For col-tile t, k-step kk, lane (half=lane>>4, n=lane&15):
//   Wp[((t*32+kk)*32+lane)*2 + j] = W[kk*4 + half*2 + j][t*16 + n]
__global__ void pack_kernel(const float* __restrict__ W, float* __restrict__ Wp, int C) {
    int gid = blockIdx.x * 256 + threadIdx.x;
    int total = (C >> 4) * 1024;           // tiles * 32 ksteps * 32 lanes
    if (gid >= total) return;
    int t    = gid >> 10;
    int kk   = (gid >> 5) & 31;
    int lane = gid & 31;
    int half = lane >> 4, n = lane & 15;
    int k0 = kk * 4 + half * 2;
    Wp[(size_t)gid * 2 + 0] = W[(size_t)k0 * C + t * 16 + n];
    Wp[(size_t)gid * 2 + 1] = W[(size_t)(k0 + 1) * C + t * 16 + n];
}

// ---------------- SAGE layer: h' = relu(BN(agg/deg @ Wl + bl + h @ Wr)) ----------------
// block = 80 nodes (5 row-tiles), 8 waves = 8 col-tiles; async-to-LDS staging,
// packed b64 weight fragments reused across 5 accumulators.
__global__ void __launch_bounds__(256)
sage_layer_kernel(const float* __restrict__ hin, const float* __restrict__ agg,
                  const float* __restrict__ deg,
                  const float* __restrict__ Wlp, const float* __restrict__ bl,
                  const float* __restrict__ Wrp,
                  const float* __restrict__ gamma, const float* __restrict__ beta,
                  const float* __restrict__ rmean, const float* __restrict__ rvar,
                  float* __restrict__ hout) {
    __shared__ float sAgg[80][132];   // padded rows: conflict-free strided A reads
    __shared__ float sH[80][132];
    __shared__ float sInv[80];

    const int tid  = threadIdx.x;
    const int wave = tid >> 5;
    const int lane = tid & 31;
    const long rowbase = (long)blockIdx.x * 80;

    // ---- async staging: 80x128 floats per array = 2560 float4 chunks ----
    const float* gA = agg + rowbase * Dd;
    const float* gH = hin + rowbase * Dd;
    #pragma unroll
    for (int i = 0; i < 10; ++i) {
        int fidx = tid + i * 256;          // float4 index: r = fidx>>5, c4 = fidx&31
        int r = fidx >> 5, c4 = fidx & 31;
        unsigned voff = (unsigned)(fidx * 16);          // = r*512 + c4*16 bytes
        lds_fp lA = (lds_fp)&sAgg[r][c4 * 4];
        lds_fp lH = (lds_fp)&sH[r][c4 * 4];
        asm volatile("global_load_async_to_lds_b128 %0, %1, %2"
                     :: "v"(lA), "v"(voff), "s"(gA) : "memory");
        asm volatile("global_load_async_to_lds_b128 %0, %1, %2"
                     :: "v"(lH), "v"(voff), "s"(gH) : "memory");
    }
    if (tid < 80) sInv[tid] = 1.0f / fmaxf(deg[rowbase + tid], 1.0f);
    asm volatile("s_wait_asynccnt 0x0" ::: "memory");
    __syncthreads();

    const int m    = lane & 15;
    const int n    = lane & 15;
    const int ksel = (lane >> 4) << 1;     // lanes 0-15 -> K=0,1 ; 16-31 -> K=2,3

    const float2* pWl = (const float2*)(Wlp + (size_t)wave * 2048);
    const float2* pWr = (const float2*)(Wrp + (size_t)wave * 2048);

    float inv[5];
    #pragma unroll
    for (int rt = 0; rt < 5; ++rt) inv[rt] = sInv[rt * 16 + m];

    v8f acc[5] = {};

    #pragma unroll 2
    for (int kk = 0; kk < 32; ++kk) {
        const float2 wl = pWl[kk * 32 + lane];
        const float2 wr = pWr[kk * 32 + lane];
        v2f bL; bL.x = wl.x; bL.y = wl.y;
        v2f bR; bR.x = wr.x; bR.y = wr.y;
        const int k0 = kk * 4 + ksel;
        #pragma unroll
        for (int rt = 0; rt < 5; ++rt) {
            const int row = rt * 16 + m;
            v2f aA, aH;
            aA.x = sAgg[row][k0] * inv[rt];  aA.y = sAgg[row][k0 + 1] * inv[rt];
            aH.x = sH[row][k0];              aH.y = sH[row][k0 + 1];
            acc[rt] = __builtin_amdgcn_wmma_f32_16x16x4_f32(false, aA, false, bL, (short)0, acc[rt], false, false);
            acc[rt] = __builtin_amdgcn_wmma_f32_16x16x4_f32(false, aH, false, bR, (short)0, acc[rt], false, false);
        }
    }

    const int col     = wave * 16 + n;
    const float scale = rsqrtf(rvar[col] + EPSf) * gamma[col];
    const float mu    = rmean[col];
    const float bb    = beta[col];
    const float bias  = bl[col];
    const int moff    = (lane >> 4) << 3;   // C/D: lanes 16-31 hold M = r+8

    #pragma unroll
    for (int rt = 0; rt < 5; ++rt) {
        #pragma unroll
        for (int r = 0; r < 8; ++r) {
            long grow = rowbase + rt * 16 + r + moff;
            float v = acc[rt][r] + bias;
            v = (v - mu) * scale + bb;
            hout[grow * Dd + col] = fmaxf(v, 0.0f);
        }
    }
}

// ---------------- global mean-pool (sum + count) ----------------
__global__ void pool_kernel(const float* __restrict__ h, const int* __restrict__ bidx,
                            float* __restrict__ gsum, float* __restrict__ gcnt) {
    long gid = (long)blockIdx.x * 256 + threadIdx.x;
    int node = (int)(gid >> 5);
    int lane = (int)(gid & 31);
    if (node >= Nn) return;
    int g = bidx[node];
    const float4 v = *reinterpret_cast<const float4*>(h + (long)node * Dd + lane * 4);
    float* p = gsum + (long)g * Dd + lane * 4;
    atomicAdd(p + 0, v.x); atomicAdd(p + 1, v.y);
    atomicAdd(p + 2, v.z); atomicAdd(p + 3, v.w);
    if (lane == 0) atomicAdd(&gcnt[g], 1.0f);
}

// ---------------- MLP layer 1: t = relu((gsum/cnt) @ W1 + b1) ----------------
__global__ void __launch_bounds__(256)
mlp1_kernel(const float* __restrict__ gsum, const float* __restrict__ gcnt,
            const float* __restrict__ W1p, const float* __restrict__ b1,
            float* __restrict__ t) {
    __shared__ float sG[16][132];
    __shared__ float sInv[16];
    const int tid = threadIdx.x, wave = tid >> 5, lane = tid & 31;
    const long rowbase = (long)blockIdx.x * 16;
    #pragma unroll
    for (int i = 0; i < 8; ++i) {
        int idx = tid + i * 256;
        int r = idx >> 7, c = idx & 127;
        sG[r][c] = gsum[(rowbase + r) * Dd + c];
    }
    if (tid < 16) sInv[tid] = 1.0f / fmaxf(gcnt[rowbase + tid], 1.0f);
    __syncthreads();

    const int m = lane & 15, n = lane & 15;
    const int ksel = (lane >> 4) << 1;
    const float inv = sInv[m];
    const float2* pW = (const float2*)(W1p + (size_t)wave * 2048);
    v8f acc = {};

    #pragma unroll 4
    for (int kk = 0; kk < 32; ++kk) {
        const float2 w = pW[kk * 32 + lane];
        v2f a, b;
        b.x = w.x; b.y = w.y;
        const int k0 = kk * 4 + ksel;
        a.x = sG[m][k0] * inv;  a.y = sG[m][k0 + 1] * inv;
        acc = __builtin_amdgcn_wmma_f32_16x16x4_f32(false, a, false, b, (short)0, acc, false, false);
    }

    const int col = wave * 16 + n;
    const float bias = b1[col];
    const int moff = (lane >> 4) << 3;
    #pragma unroll
    for (int r = 0; r < 8; ++r) {
        long grow = rowbase + r + moff;
        t[grow * Dd + col] = fmaxf(acc[r] + bias, 0.0f);
    }
}

// ---------------- MLP layer 2: out = t @ W2 + b2  ([512,128]@[128,64]) ----------------
__global__ void __launch_bounds__(128)
mlp2_kernel(const float* __restrict__ t, const float* __restrict__ W2p,
            const float* __restrict__ b2, float* __restrict__ out) {
    __shared__ float sT[16][132];
    const int tid = threadIdx.x, wave = tid >> 5, lane = tid & 31;
    const long rowbase = (long)blockIdx.x * 16;
    #pragma unroll
    for (int i = 0; i < 16; ++i) {
        int idx = tid + i * 128;
        int r = idx >> 7, c = idx & 127;
        sT[r][c] = t[(rowbase + r) * Dd + c];
    }
    __syncthreads();

    const int m = lane & 15, n = lane & 15;
    const int ksel = (lane >> 4) << 1;
    const float2* pW = (const float2*)(W2p + (size_t)wave * 2048);
    v8f acc = {};

    #pragma unroll 4
    for (int kk = 0; kk < 32; ++kk) {
        const float2 w = pW[kk * 32 + lane];
        v2f a, b;
        b.x = w.x; b.y = w.y;
        const int k0 = kk * 4 + ksel;
        a.x = sT[m][k0];  a.y = sT[m][k0 + 1];
        acc = __builtin_amdgcn_wmma_f32_16x16x4_f32(false, a, false, b, (short)0, acc, false, false);
    }

    const int col = wave * 16 + n;
    const float bias = b2[col];
    const int moff = (lane >> 4) << 3;
    #pragma unroll
    for (int r = 0; r < 8; ++r) {
        long grow = rowbase + r + moff;
        out[grow * OUTd + col] = acc[r] + bias;
    }
}

extern "C" void kernel_launch(void* const* d_in, const int* in_sizes, int n_in,
                              void* d_out, int out_size, void* d_ws, size_t ws_size,
                              hipStream_t stream) {
    const float* x     = (const float*)d_in[0];
    const int*   ei    = (const int*)  d_in[1];
    const int*   bidx  = (const int*)  d_in[2];
    const float* Wl    = (const float*)d_in[3];
    const float* bl    = (const float*)d_in[4];
    const float* Wr    = (const float*)d_in[5];
    const float* gamma = (const float*)d_in[6];
    const float* beta  = (const float*)d_in[7];
    const float* rmean = (const float*)d_in[8];
    const float* rvar  = (const float*)d_in[9];
    const float* W1    = (const float*)d_in[10];
    const float* b1    = (const float*)d_in[11];
    const float* W2    = (const float*)d_in[12];
    const float* b2    = (const float*)d_in[13];
    float* out = (float*)d_out;

    const int* src = ei;        // ei[0, :]
    const int* dst = ei + Ee;   // ei[1, :]

    char* ws = (char*)d_ws;
    size_t o = 0;
    float* h0   = (float*)(ws + o); o += (size_t)Nn * Dd * 4;
    float* h1   = (float*)(ws + o); o += (size_t)Nn * Dd * 4;
    float* agg  = (float*)(ws + o); o += (size_t)Nn * Dd * 4;
    float* deg  = (float*)(ws + o); o += (size_t)Nn * 4;
    float* gsum = (float*)(ws + o); o += (size_t)Gg * Dd * 4;
    float* gcnt = (float*)(ws + o); o += (size_t)Gg * 4;
    float* tb   = (float*)(ws + o); o += (size_t)Gg * Dd * 4;
    float* Wlp  = (float*)(ws + o); o += (size_t)3 * Dd * Dd * 4;
    float* Wrp  = (float*)(ws + o); o += (size_t)3 * Dd * Dd * 4;
    float* W1p  = (float*)(ws + o); o += (size_t)Dd * Dd * 4;
    float* W2p  = (float*)(ws + o); o += (size_t)Dd * OUTd * 4;

    // pack weights into WMMA B-fragment order (b64 per lane per k-step)
    for (int l = 0; l < 3; ++l) {
        pack_kernel<<<32, 256, 0, stream>>>(Wl + (size_t)l * Dd * Dd, Wlp + (size_t)l * Dd * Dd, Dd);
        pack_kernel<<<32, 256, 0, stream>>>(Wr + (size_t)l * Dd * Dd, Wrp + (size_t)l * Dd * Dd, Dd);
    }
    pack_kernel<<<32, 256, 0, stream>>>(W1, W1p, Dd);
    pack_kernel<<<16, 256, 0, stream>>>(W2, W2p, OUTd);

    hipMemsetAsync(deg, 0, (size_t)Nn * 4, stream);
    deg_kernel<<<(Ee + 255) / 256, 256, 0, stream>>>(dst, deg);

    const float* hin = x;
    float* houts[3] = {h0, h1, h0};
    for (int l = 0; l < 3; ++l) {
        hipMemsetAsync(agg, 0, (size_t)Nn * Dd * 4, stream);
        scatter_kernel<<<(int)(((long)Ee * 32) / 256), 256, 0, stream>>>(hin, src, dst, agg);
        sage_layer_kernel<<<Nn / 80, 256, 0, stream>>>(
            hin, agg, deg,
            Wlp + (size_t)l * Dd * Dd, bl + (size_t)l * Dd, Wrp + (size_t)l * Dd * Dd,
            gamma + (size_t)l * Dd, beta + (size_t)l * Dd,
            rmean + (size_t)l * Dd, rvar + (size_t)l * Dd,
            houts[l]);
        hin = houts[l];
    }

    hipMemsetAsync(gsum, 0, (size_t)Gg * Dd * 4, stream);
    hipMemsetAsync(gcnt, 0, (size_t)Gg * 4, stream);
    pool_kernel<<<(int)(((long)Nn * 32 + 255) / 256), 256, 0, stream>>>(hin, bidx, gsum, gcnt);

    mlp1_kernel<<<Gg / 16, 256, 0, stream>>>(gsum, gcnt, W1p, b1, tb);
    mlp2_kernel<<<Gg / 16, 128, 0, stream>>>(tb, W2p, b2, out);
}